// Faithful_SAE_85409719649036
// MI455X (gfx1250) — compile-verified
//
#include <hip/hip_runtime.h>
#include <hip/hip_bf16.h>

// ---------------------------------------------------------------------------
// Faithful SAE forward on MI455X (gfx1250), wave32.
//   latent = x @ encoder        (fp32 WMMA 16x16x4 -> exact fp32 math)
//   sparse = topk32(latent)     (radix select per row, in-place zeroing)
//   h      = sparse @ decoder   (gather: 32 weighted decoder rows per output row)
// d_out = [h (4096*768) | sparse (4096*16384)]  (reference returns (h, sparse))
// ---------------------------------------------------------------------------

#define BATCH      4096
#define IN_DIM     768
#define LATENT     16384
#define HID_DIM    768
#define TOPK       32

typedef __attribute__((ext_vector_type(2))) float v2f;
typedef __attribute__((ext_vector_type(8))) float v8f;

// ------------------------- Kernel 1: encode GEMM ---------------------------
// Macro tile 32(M) x 64(N), 256 threads = 8 waves, wave -> one 16x16 tile.
// K staged in chunks of 32 through LDS (padded strides vs bank conflicts).
#define MTILE 32
#define NTILE 64
#define KCHUNK 32
#define A_STRIDE 36   // 32 + 4 pad, keeps 16B alignment
#define B_STRIDE 68   // 64 + 4 pad, keeps 16B alignment

__global__ __launch_bounds__(256) void encode_gemm(const float* __restrict__ x,
                                                   const float* __restrict__ enc,
                                                   float* __restrict__ latent) {
    __shared__ __align__(16) float As[MTILE * A_STRIDE];
    __shared__ __align__(16) float Bs[KCHUNK * B_STRIDE];

    const int tid   = threadIdx.x;
    const int lane  = tid & 31;          // wave32
    const int wave  = tid >> 5;          // 0..7
    const int wm    = wave >> 2;         // 0..1  (16-row slab)
    const int wn    = wave & 3;          // 0..3  (16-col slab)
    const int hsel  = lane >> 4;         // 0/1 -> K pair select per ISA layout
    const int lrow  = lane & 15;

    const int m0 = blockIdx.y * MTILE;
    const int n0 = blockIdx.x * NTILE;

    // Staging maps (fully coalesced float4):
    const int a_r  = tid >> 3;           // 0..31
    const int a_c  = (tid & 7) * 4;      // 0..28
    const int b_r  = tid >> 3;           // 0..31
    const int b_c  = (tid & 7) * 8;      // 0..56  (two float4 per thread)

    v8f acc = {};                        // fp32 C/D accumulator (8 VGPRs)

    for (int k0 = 0; k0 < IN_DIM; k0 += KCHUNK) {
        // A chunk: x[m0..m0+31, k0..k0+31]
        const float4 av = *(const float4*)(x + (size_t)(m0 + a_r) * IN_DIM + k0 + a_c);
        *(float4*)(&As[a_r * A_STRIDE + a_c]) = av;
        // B chunk: enc[k0..k0+31, n0..n0+63]
        const float4 bv0 = *(const float4*)(enc + (size_t)(k0 + b_r) * LATENT + n0 + b_c);
        const float4 bv1 = *(const float4*)(enc + (size_t)(k0 + b_r) * LATENT + n0 + b_c + 4);
        *(float4*)(&Bs[b_r * B_STRIDE + b_c])     = bv0;
        *(float4*)(&Bs[b_r * B_STRIDE + b_c + 4]) = bv1;
        __syncthreads();

        // Hint next A chunk into cache while we crunch this one.
        if (k0 + KCHUNK < IN_DIM)
            __builtin_prefetch(x + (size_t)(m0 + a_r) * IN_DIM + k0 + KCHUNK + a_c, 0, 3);

#pragma unroll
        for (int kk = 0; kk < KCHUNK; kk += 4) {
            // A frag 16x4: lanes 0-15 hold K={0,1}, lanes 16-31 K={2,3}
            v2f a, b;
            const int am = (wm * 16 + lrow) * A_STRIDE + kk + 2 * hsel;
            a.x = As[am];
            a.y = As[am + 1];
            // B frag 4x16: rows striped across lanes within a VGPR
            const int bn = wn * 16 + lrow;
            b.x = Bs[(kk + 2 * hsel) * B_STRIDE + bn];
            b.y = Bs[(kk + 2 * hsel + 1) * B_STRIDE + bn];
            acc = __builtin_amdgcn_wmma_f32_16x16x4_f32(
                false, a, false, b, (short)0, acc, false, false);
        }
        __syncthreads();
    }

    // C/D layout: VGPR r, lane l -> M = r + 8*(l>=16), N = l&15
    const int nglob = n0 + wn * 16 + lrow;
#pragma unroll
    for (int r = 0; r < 8; ++r) {
        const int mglob = m0 + wm * 16 + r + 8 * hsel;
        latent[(size_t)mglob * LATENT + nglob] = acc[r];
    }
}

// ------------------------- Kernel 2: top-K in place ------------------------
// One 256-thread block per row. Exact K-th-largest via 4x8-bit radix select
// on order-preserving keys; ties broken by lowest index (matches top_k).
// Kept entries stay, others zeroed; compact (idx,val) lists -> workspace.
#define ROW_TPB 256
#define CHUNK   (LATENT / ROW_TPB)   // 64 contiguous elements per thread

__device__ __forceinline__ unsigned int f2key(float f) {
    unsigned int u = __float_as_uint(f);
    return (u & 0x80000000u) ? ~u : (u | 0x80000000u);
}

__global__ __launch_bounds__(ROW_TPB) void topk_inplace(float* __restrict__ sparse,
                                                        int* __restrict__ wsIdx,
                                                        float* __restrict__ wsVal) {
    const int row = blockIdx.x;
    float* rp = sparse + (size_t)row * LATENT;
    const int tid  = threadIdx.x;
    const int base = tid * CHUNK;

    __shared__ unsigned int hist[256];
    __shared__ unsigned int cntG[ROW_TPB], cntE[ROW_TPB];
    __shared__ unsigned int eqPre[ROW_TPB], keptPre[ROW_TPB];
    __shared__ unsigned int sPrefix, sNeed, sE;

    if (tid == 0) { sPrefix = 0u; sNeed = TOPK; }
    __syncthreads();

    // ---- radix select: find the K-th largest key ----
    for (int shift = 24; shift >= 0; shift -= 8) {
        hist[tid] = 0u;
        __syncthreads();
        const unsigned int prefix = sPrefix;
        const unsigned int hiMask = (shift == 24) ? 0u : (0xFFFFFFFFu << (shift + 8));
        for (int i = 0; i < CHUNK; i += 4) {
            const float4 f4 = *(const float4*)(rp + base + i);
            const float fs[4] = {f4.x, f4.y, f4.z, f4.w};
#pragma unroll
            for (int c = 0; c < 4; ++c) {
                const unsigned int key = f2key(fs[c]);
                if (((key ^ prefix) & hiMask) == 0u)
                    atomicAdd(&hist[(key >> shift) & 255u], 1u);
            }
        }
        __syncthreads();
        if (tid == 0) {
            unsigned int need = sNeed, cum = 0u;
            for (int d = 255; d >= 0; --d) {
                const unsigned int c = hist[d];
                if (cum + c >= need || d == 0) {
                    sNeed   = need - cum;
                    sPrefix = prefix | ((unsigned int)d << shift);
                    break;
                }
                cum += c;
            }
        }
        __syncthreads();
    }
    const unsigned int T = sPrefix;   // key of the K-th largest element

    // ---- pass A: per-thread counts of (>T) and (==T) over contiguous chunk ----
    unsigned int g = 0u, e = 0u;
    for (int i = 0; i < CHUNK; i += 4) {
        const float4 f4 = *(const float4*)(rp + base + i);
        const float fs[4] = {f4.x, f4.y, f4.z, f4.w};
#pragma unroll
        for (int c = 0; c < 4; ++c) {
            const unsigned int key = f2key(fs[c]);
            g += (key > T);
            e += (key == T);
        }
    }
    cntG[tid] = g; cntE[tid] = e;
    __syncthreads();

    // ---- serial prefix (256 entries, thread 0): deterministic slots + tie ranks ----
    if (tid == 0) {
        unsigned int G = 0u;
        for (int t = 0; t < ROW_TPB; ++t) G += cntG[t];
        int Ei = (int)TOPK - (int)G;           // #ties at T to keep (>=1 by construction)
        if (Ei < 0) Ei = 0;
        const unsigned int E = (unsigned int)Ei;
        sE = E;
        unsigned int eq = 0u, kp = 0u;
        for (int t = 0; t < ROW_TPB; ++t) {
            eqPre[t]   = eq;
            keptPre[t] = kp;
            const unsigned int room   = (eq < E) ? (E - eq) : 0u;
            const unsigned int keptEq = (cntE[t] < room) ? cntE[t] : room;
            kp += cntG[t] + keptEq;
            eq += cntE[t];
        }
    }
    __syncthreads();

    // ---- pass B: zero non-kept, emit compact (idx,val) deterministically ----
    const unsigned int E = sE;
    unsigned int eqSeen = eqPre[tid];
    unsigned int slot   = keptPre[tid];
    int*   rIdx = wsIdx + (size_t)row * TOPK;
    float* rVal = wsVal + (size_t)row * TOPK;
    for (int i = 0; i < CHUNK; ++i) {
        const int j = base + i;
        const float v = rp[j];
        const unsigned int key = f2key(v);
        bool kept = false;
        if (key > T) kept = true;
        else if (key == T) { kept = (eqSeen < E); ++eqSeen; }
        if (kept) {
            if (slot < TOPK) { rIdx[slot] = j; rVal[slot] = v; }
            ++slot;
        } else {
            rp[j] = 0.0f;
        }
    }
}

// ------------------------- Kernel 3: decode gather -------------------------
// h[row,:] = sum_k val[k] * decoder[idx[k], :]. Decoder (48 MB) is L2-hot.
__global__ __launch_bounds__(256) void decode_gather(const float* __restrict__ dec,
                                                     const int* __restrict__ wsIdx,
                                                     const float* __restrict__ wsVal,
                                                     float* __restrict__ h) {
    const int row = blockIdx.x;
    const int tid = threadIdx.x;
    __shared__ int   sIdx[TOPK];
    __shared__ float sVal[TOPK];
    if (tid < TOPK) {
        sIdx[tid] = wsIdx[(size_t)row * TOPK + tid];
        sVal[tid] = wsVal[(size_t)row * TOPK + tid];
    }
    __syncthreads();

    float a0 = 0.0f, a1 = 0.0f, a2 = 0.0f;   // 768 = 3 * 256 columns
#pragma unroll 4
    for (int k = 0; k < TOPK; ++k) {
        const float v = sVal[k];
        const float* drow = dec + (size_t)sIdx[k] * HID_DIM;
        a0 = fmaf(v, drow[tid],       a0);
        a1 = fmaf(v, drow[tid + 256], a1);
        a2 = fmaf(v, drow[tid + 512], a2);
    }
    float* hr = h + (size_t)row * HID_DIM;
    hr[tid]       = a0;
    hr[tid + 256] = a1;
    hr[tid + 512] = a2;
}

// ------------------------------- launcher ----------------------------------
extern "C" void kernel_launch(void* const* d_in, const int* in_sizes, int n_in,
                              void* d_out, int out_size, void* d_ws, size_t ws_size,
                              hipStream_t stream) {
    const float* x   = (const float*)d_in[0];   // [4096, 768]
    const float* enc = (const float*)d_in[1];   // [768, 16384]
    const float* dec = (const float*)d_in[2];   // [16384, 768]

    float* h_out      = (float*)d_out;                              // [4096, 768]
    float* sparse_out = h_out + (size_t)BATCH * HID_DIM;            // [4096, 16384]

    int*   wsIdx = (int*)d_ws;                                      // 4096*32 ints
    float* wsVal = (float*)d_ws + (size_t)BATCH * TOPK;             // 4096*32 floats

    dim3 g1(LATENT / NTILE, BATCH / MTILE);   // 256 x 128 workgroups
    encode_gemm<<<g1, 256, 0, stream>>>(x, enc, sparse_out);
    topk_inplace<<<BATCH, ROW_TPB, 0, stream>>>(sparse_out, wsIdx, wsVal);
    decode_gather<<<BATCH, 256, 0, stream>>>(dec, wsIdx, wsVal, h_out);
}